// RetinaStaticExportWrapper_76398878261254
// MI455X (gfx1250) — compile-verified
//
#include <hip/hip_runtime.h>
#include <stdint.h>

// RetinaFace post-process for MI455X (gfx1250, wave32).
// Pipeline: histogram-threshold top-750 -> bitonic exact sort -> lazy decode
// -> parallel IoU bitmask (LDS-staged boxes, wave32 ballot) -> single-wave
// greedy NMS sweep over 72KB mask held in (CDNA5-sized) dynamic LDS.
// CDNA5 paths: TDM tensor_load_to_lds + s_wait_tensorcnt (bulk LDS fill),
// global_load_async_to_lds_b128 + s_wait_asynccnt (per-lane async DMA),
// ballot_w32 / ds_bpermute (wave32), global_prefetch_b8.

#define P_NUM    420000
#define TOPK     750
#define IMG_F    3200.0f
#define VAR0_F   0.1f
#define VAR1_F   0.2f
#define CONF_THR 0.02f
#define NMS_THR  0.4f
#define BINS     4096
#define CANDN    2048
#define MASKW    24            // ceil(750/32) mask words per row

struct Cand { float s; int i; };

typedef uint32_t u32x4 __attribute__((ext_vector_type(4)));
typedef int      i32x4 __attribute__((ext_vector_type(4)));
typedef int      i32x8 __attribute__((ext_vector_type(8)));

// ---------- CDNA5 helpers ----------

// Flat LDS addresses are {SHARED_BASE[63:32], lds_offset[31:0]} (ISA 10.2),
// so truncating a flat pointer to 32 bits yields the wave-relative LDS offset
// the async-to-LDS / TDM lds_addr operands want.
__device__ __forceinline__ uint32_t lds_off_of(const void* p) {
  return (uint32_t)(uint64_t)(uintptr_t)p;
}

// Async DMA global -> LDS, 16B per lane. Tracked by ASYNCcnt.
__device__ __forceinline__ void async_ld_b128(uint32_t lds_off, const void* gaddr) {
  asm volatile("global_load_async_to_lds_b128 %0, %1, off"
               :: "v"(lds_off), "v"(gaddr) : "memory");
}
__device__ __forceinline__ void wait_async0() {
  asm volatile("s_wait_asynccnt 0" ::: "memory");
}

// Tensor Data Mover: one instruction streams nwords*4 contiguous bytes from
// global into LDS. 1-D descriptor: data_size=4B, tile_dim0=tensor_dim0=nwords
// (nwords <= 65535), groups 2/3 zeroed (<=2D tensor). Tracked by TENSORcnt.
// This toolchain's builtin is the 6-arg form:
//   (u32x4 g0, i32x8 g1, i32x4, i32x4, i32x8, i32 cpol)
__device__ __forceinline__ void tdm_load_words_to_lds(uint32_t lds_off,
                                                      const void* gsrc,
                                                      uint32_t nwords) {
  uint64_t ga = (uint64_t)(uintptr_t)gsrc;
  u32x4 g0;
  g0[0] = 1u;                                            // count=1, user mode
  g0[1] = lds_off;                                       // lds_addr (bytes)
  g0[2] = (uint32_t)ga;                                  // global_addr[31:0]
  g0[3] = ((uint32_t)(ga >> 32) & 0x01FFFFFFu)           // global_addr[56:32]
        | (2u << 30);                                    // type=2 ("image")
  i32x8 g1;
  g1[0] = (int)(2u << 16);                               // data_size=2 -> 4B
  g1[1] = (int)((nwords & 0xFFFFu) << 16);               // tensor_dim0[15:0]
  g1[2] = (int)(nwords >> 16);                           // tensor_dim0[31:16]
  g1[3] = (int)(nwords << 16);                           // tile_dim0
  g1[4] = 0;                                             // tile_dim1/2 unused
  g1[5] = (int)nwords;                                   // dim0_stride low
  g1[6] = 0;
  g1[7] = 0;
  i32x4 gz4 = {0, 0, 0, 0};
  i32x8 gz8 = {0, 0, 0, 0, 0, 0, 0, 0};
#if __has_builtin(__builtin_amdgcn_tensor_load_to_lds)
  __builtin_amdgcn_tensor_load_to_lds(g0, g1, gz4, gz4, gz8, 0);
#else
  asm volatile("tensor_load_to_lds %0, %1, %2, %3"
               :: "s"(g0), "s"(g1), "s"(gz4), "s"(gz4) : "memory");
#endif
}
__device__ __forceinline__ void wait_tensor0() {
#if __has_builtin(__builtin_amdgcn_s_wait_tensorcnt)
  __builtin_amdgcn_s_wait_tensorcnt(0);
#else
  asm volatile("s_wait_tensorcnt 0x0" ::: "memory");
#endif
}

__device__ __forceinline__ uint32_t ballot32(bool p) {
#if __has_builtin(__builtin_amdgcn_ballot_w32)
  return __builtin_amdgcn_ballot_w32(p);
#else
  return (uint32_t)__ballot(p);
#endif
}

__device__ __forceinline__ uint32_t bcast_lane(uint32_t v, int lane) {
  // wave32 broadcast via LDS-permute hardware (no LDS memory touched)
  return (uint32_t)__builtin_amdgcn_ds_bpermute(lane << 2, (int)v);
}

__device__ __forceinline__ int bin_of(float s) {
  int b = (int)(s * (float)BINS);
  if (b < 0) b = 0;
  if (b > BINS - 1) b = BINS - 1;
  return b;
}

// ---------- kernels ----------

__global__ void k_zero(int* hist, int* counter) {
  int t = blockIdx.x * blockDim.x + threadIdx.x;
  if (t < BINS) hist[t] = 0;
  if (t == 0) *counter = 0;
}

__global__ void k_hist(const float* __restrict__ conf, int* __restrict__ hist) {
  int i = blockIdx.x * blockDim.x + threadIdx.x;
  if (i >= P_NUM) return;
  __builtin_prefetch(conf + 2 * i + 4096, 0, 0);   // global_prefetch_b8
  float s = conf[2 * i + 1];
  if (s > CONF_THR) atomicAdd(&hist[bin_of(s)], 1);
}

// Find b* = max{b : sum_{j>=b} hist[j] >= TOPK} (0 if total < TOPK).
// Top-750 scores all live in bins >= b*, and count(bins >= b*) <= 749 + |bin b*|.
__global__ void k_thr(const int* __restrict__ hist, int* thr_bin, int* counter) {
  __shared__ int chunk[256];
  const int CW = BINS / 256; // 16
  int t = threadIdx.x;
  int sum = 0;
  for (int r = 0; r < CW; ++r) sum += hist[t * CW + r];
  chunk[t] = sum;
  __syncthreads();
  if (t == 0) {
    int acc = 0, b = 0;
    for (int c = 255; c >= 0; --c) {
      if (acc + chunk[c] >= TOPK) {
        for (int bb = c * CW + CW - 1; bb >= c * CW; --bb) {
          acc += hist[bb];
          if (acc >= TOPK) { b = bb; break; }
        }
        break;
      }
      acc += chunk[c];
    }
    *thr_bin = b;
    *counter = 0;
  }
}

__global__ void k_compact(const float* __restrict__ conf,
                          const int* __restrict__ thr_bin,
                          int* counter, Cand* __restrict__ cand) {
  int i = blockIdx.x * blockDim.x + threadIdx.x;
  if (i >= P_NUM) return;
  float s = conf[2 * i + 1];
  if (s > CONF_THR && bin_of(s) >= *thr_bin) {
    int pos = atomicAdd(counter, 1);
    if (pos < CANDN) { cand[pos].s = s; cand[pos].i = i; }
  }
}

__device__ __forceinline__ bool cand_before(Cand a, Cand b) {
  // exact top_k order: descending score, ascending index on ties
  return (a.s > b.s) || (a.s == b.s && a.i < b.i);
}

__global__ void k_sort_decode(const Cand* __restrict__ cand, const int* __restrict__ counter,
                              const float* __restrict__ loc, const float* __restrict__ landms,
                              const float* __restrict__ prior,
                              float* __restrict__ scores_s, float* __restrict__ boxes_s,
                              float* __restrict__ lm_s) {
  __shared__ __align__(16) Cand sh[CANDN];
  int t = threadIdx.x; // 1024 threads
  // TDM: one DMA descriptor streams the 16KB candidate buffer into LDS.
  if (t < 32) {
    tdm_load_words_to_lds(lds_off_of(sh), (const void*)cand, CANDN * 2);
    wait_tensor0();
  }
  __syncthreads();
  int cnt = *counter;
  if (cnt > CANDN) cnt = CANDN;
  // Pad slots beyond cnt so they sink to the bottom.
  for (int s0 = t; s0 < CANDN; s0 += 1024)
    if (s0 >= cnt) { sh[s0].s = -1.0f; sh[s0].i = 0x7FFFFFFF; }
  __syncthreads();
  // Bitonic sort, 2048 elements, 1024 compare-exchanges per step.
  for (int k = 2; k <= CANDN; k <<= 1) {
    for (int j = k >> 1; j > 0; j >>= 1) {
      int i = ((t & ~(j - 1)) << 1) | (t & (j - 1));
      int l = i | j;
      bool up = ((i & k) == 0);
      Cand a = sh[i], b = sh[l];
      if (cand_before(b, a) == up) { sh[i] = b; sh[l] = a; }
      __syncthreads();
    }
  }
  // Decode only the surviving 750.
  if (t < TOPK) {
    Cand c = sh[t];
    if (c.s > CONF_THR) {
      int idx = c.i;
      float4 p = ((const float4*)prior)[idx];
      float4 l = ((const float4*)loc)[idx];
      float cx = p.x + l.x * VAR0_F * p.z;
      float cy = p.y + l.y * VAR0_F * p.w;
      float w  = p.z * expf(l.z * VAR1_F);
      float h  = p.w * expf(l.w * VAR1_F);
      float x1 = cx - 0.5f * w, y1 = cy - 0.5f * h;
      boxes_s[t * 4 + 0] = x1 * IMG_F;
      boxes_s[t * 4 + 1] = y1 * IMG_F;
      boxes_s[t * 4 + 2] = (x1 + w) * IMG_F;
      boxes_s[t * 4 + 3] = (y1 + h) * IMG_F;
      #pragma unroll
      for (int j = 0; j < 5; ++j) {
        float2 lv = ((const float2*)landms)[idx * 5 + j];
        lm_s[t * 10 + 2 * j + 0] = (p.x + p.z * lv.x * VAR0_F) * IMG_F;
        lm_s[t * 10 + 2 * j + 1] = (p.y + p.w * lv.y * VAR0_F) * IMG_F;
      }
      scores_s[t] = c.s;
    } else {
      scores_s[t] = -1.0f;
      #pragma unroll
      for (int j = 0; j < 4; ++j)  boxes_s[t * 4 + j] = 0.0f;
      #pragma unroll
      for (int j = 0; j < 10; ++j) lm_s[t * 10 + j] = 0.0f;
    }
  }
}

// mask[i][bit j] = (iou(i,j) > thr) && (j > i). 750 blocks, boxes staged via
// per-lane async DMA (ASYNCcnt path).
__global__ void k_iou(const float* __restrict__ boxes_s, uint32_t* __restrict__ mask) {
  __shared__ __align__(16) float sb[TOPK * 4];
  int t = threadIdx.x; // 256
  for (int c = t; c < TOPK; c += 256)
    async_ld_b128(lds_off_of(&sb[c * 4]), (const void*)(boxes_s + c * 4));
  wait_async0();
  __syncthreads();
  int i = blockIdx.x;
  float4 bi = *(const float4*)&sb[i * 4];
  float ai = (bi.z - bi.x) * (bi.w - bi.y);
  for (int base = 0; base < MASKW * 32; base += 256) {
    int j = base + t;
    bool pred = false;
    if (j < TOPK && j > i) {
      float4 bj = *(const float4*)&sb[j * 4];
      float aj = (bj.z - bj.x) * (bj.w - bj.y);
      float lx = fmaxf(bi.x, bj.x), ly = fmaxf(bi.y, bj.y);
      float rx = fminf(bi.z, bj.z), ry = fminf(bi.w, bj.w);
      float wd = fmaxf(rx - lx, 0.0f), hg = fmaxf(ry - ly, 0.0f);
      float inter = wd * hg;
      float iou = inter / (ai + aj - inter);
      pred = iou > NMS_THR;
    }
    uint32_t bl = ballot32(pred);
    if ((t & 31) == 0) mask[i * MASKW + (j >> 5)] = bl;
  }
}

// Single-wave greedy sweep: 24 mask words held one per lane; the 72KB mask is
// streamed into dynamic LDS by one TDM descriptor (needs CDNA5's 320KB WGP LDS),
// overlapped with the score copy. Then masked write-out of all 11250 outputs.
__global__ void k_nms(const float* __restrict__ scores_s, const float* __restrict__ boxes_s,
                      const float* __restrict__ lm_s, const uint32_t* __restrict__ mask,
                      float* __restrict__ out) {
  extern __shared__ uint32_t sm[];
  uint32_t* smask  = sm;                           // 18000 words
  float*    sscore = (float*)&sm[TOPK * MASKW];    // 750
  uint32_t* skeep  = &sm[TOPK * MASKW + TOPK];     // 24
  int t = threadIdx.x; // 1024
  if (t < 32)
    tdm_load_words_to_lds(lds_off_of(smask), (const void*)mask, TOPK * MASKW);
  for (int c = t; c < TOPK; c += 1024) sscore[c] = scores_s[c];  // overlaps DMA
  if (t < 32) wait_tensor0();
  __syncthreads();
  if (t < 32) {
    uint32_t suppw = 0, keepw = 0;
    int lane = t;
    for (int i = 0; i < TOPK; ++i) {
      int wi = i >> 5, bi = i & 31;
      uint32_t sw = bcast_lane(suppw, wi);
      bool suppressed = (sw >> bi) & 1u;
      bool alive = (sscore[i] > CONF_THR) && !suppressed; // uniform across wave
      if (alive) {
        if (lane == wi) keepw |= (1u << bi);
        if (lane < MASKW) suppw |= smask[i * MASKW + lane];
      }
    }
    if (lane < MASKW) skeep[lane] = keepw;
  }
  __syncthreads();
  for (int r = t; r < TOPK; r += 1024) {
    float kf = ((skeep[r >> 5] >> (r & 31)) & 1u) ? 1.0f : 0.0f;
    out[r] = sscore[r] * kf;
    #pragma unroll
    for (int c = 0; c < 4; ++c)  out[TOPK + r * 4 + c]      = boxes_s[r * 4 + c] * kf;
    #pragma unroll
    for (int c = 0; c < 10; ++c) out[TOPK * 5 + r * 10 + c] = lm_s[r * 10 + c] * kf;
  }
}

// ---------- launch ----------

extern "C" void kernel_launch(void* const* d_in, const int* in_sizes, int n_in,
                              void* d_out, int out_size, void* d_ws, size_t ws_size,
                              hipStream_t stream) {
  (void)in_sizes; (void)n_in; (void)out_size; (void)ws_size;
  const float* loc    = (const float*)d_in[1];
  const float* conf   = (const float*)d_in[2];
  const float* landms = (const float*)d_in[3];
  const float* prior  = (const float*)d_in[4];

  char* ws = (char*)d_ws;                        // total ws use: ~150 KB
  int*      hist     = (int*)(ws + 0);           // 16384 B
  int*      thr_bin  = (int*)(ws + 16384);
  int*      counter  = (int*)(ws + 16388);
  Cand*     cand     = (Cand*)(ws + 16400);      // 16384 B, 16B aligned
  float*    scores_s = (float*)(ws + 32800);     // 3000 B
  float*    boxes_s  = (float*)(ws + 35808);     // 12000 B, 16B aligned
  float*    lm_s     = (float*)(ws + 47808);     // 30000 B
  uint32_t* mask     = (uint32_t*)(ws + 77808);  // 72000 B, 16B aligned
  float*    out      = (float*)d_out;

  int nb = (P_NUM + 255) / 256;
  k_zero<<<BINS / 256, 256, 0, stream>>>(hist, counter);
  k_hist<<<nb, 256, 0, stream>>>(conf, hist);
  k_thr<<<1, 256, 0, stream>>>(hist, thr_bin, counter);
  k_compact<<<nb, 256, 0, stream>>>(conf, thr_bin, counter, cand);
  k_sort_decode<<<1, 1024, 0, stream>>>(cand, counter, loc, landms, prior,
                                        scores_s, boxes_s, lm_s);
  k_iou<<<TOPK, 256, 0, stream>>>(boxes_s, mask);
  const uint32_t smem = (TOPK * MASKW + TOPK + MASKW) * 4 + 40; // ~75 KB (CDNA5 LDS)
  k_nms<<<1, 1024, smem, stream>>>(scores_s, boxes_s, lm_s, mask, out);
}